// RoIAlign_2104533975273
// MI455X (gfx1250) — compile-verified
//
#include <hip/hip_runtime.h>
#include <hip/hip_bf16.h>
#include <stdint.h>

typedef __attribute__((ext_vector_type(2))) float v2f;
typedef __attribute__((ext_vector_type(8))) float v8f;

#define AH 7
#define AW 7
#define C_DIM 256
#define T_DIM 8
#define H_DIM 64
#define W_DIM 64
#define SCALE 0.0625f
#define SPLIT 8                         // workgroups per ROI
#define SLICES (C_DIM * T_DIM)          // 2048 (c,t) slices per ROI
#define SLICES_PER_WG (SLICES / SPLIT)  // 256
#define WAVES 8
#define ITERS (SLICES_PER_WG / WAVES)   // 32 slices per wave

// OUT(7x7) = Wy(16x16) * [ G(16x16) * Wx(16x16) ]
// G: gathered feature tile; row slot 2i->y0[i], 2i+1->y0[i]+1 (clamped);
//    col slot 2j->x0[j], 2j+1->x0[j]+1 (clamped). Unused slots zero-weighted.
// GEMM2's K index is PERMUTED so that the D-layout accumulator of GEMM1 is
// directly consumable as GEMM2's B fragments (no cross-lane shuffles):
//   perm(4*kc + p) = 2*kc + (p&1) + 8*(p>>1)   ->  B chunk kc = (h[2kc], h[2kc+1])
// Wy is therefore built with columns k0 = (py<4 ? 4*py : 4*(py-4)+2), k0+1.
__global__ __launch_bounds__(256) void roi_align_wmma_kernel(
    const float* __restrict__ features,
    const float* __restrict__ rois,
    float* __restrict__ out)
{
    __shared__ float sWx[256];   // 16x16 row-major [colslot][px]
    __shared__ float sWy[256];   // 16x16 row-major [py][k_perm]
    __shared__ int   sRow[16];
    __shared__ int   sCol[16];

    const int tid  = threadIdx.x;
    const int wg   = blockIdx.x;
    const int r    = wg / SPLIT;
    const int part = wg % SPLIT;

    // ---- phase 1: clear weight matrices / index slots ----
    sWx[tid] = 0.0f;
    sWy[tid] = 0.0f;
    if (tid < 16) { sRow[tid] = 0; sCol[tid] = 0; }
    __syncthreads();

    const float rb  = rois[r * 5 + 0];
    const float rx1 = rois[r * 5 + 1] * SCALE;
    const float ry1 = rois[r * 5 + 2] * SCALE;
    const float rx2 = rois[r * 5 + 3] * SCALE;
    const float ry2 = rois[r * 5 + 4] * SCALE;

    if (tid < AW) {
        const float bw = fmaxf(rx2 - rx1, 1.0f) * (1.0f / (float)AW);
        const float bh = fmaxf(ry2 - ry1, 1.0f) * (1.0f / (float)AH);
        float xs = rx1 + ((float)tid + 0.5f) * bw;
        float ys = ry1 + ((float)tid + 0.5f) * bh;
        xs = fminf(fmaxf(xs, 0.0f), (float)(W_DIM - 1));
        ys = fminf(fmaxf(ys, 0.0f), (float)(H_DIM - 1));
        const int   x0  = (int)floorf(xs);
        const int   y0  = (int)floorf(ys);
        const float wx  = xs - (float)x0;
        const float wy  = ys - (float)y0;
        const int   x1i = min(x0 + 1, W_DIM - 1);
        const int   y1i = min(y0 + 1, H_DIM - 1);
        sCol[2 * tid]     = x0;
        sCol[2 * tid + 1] = x1i;
        sRow[2 * tid]     = y0;
        sRow[2 * tid + 1] = y1i;
        sWx[(2 * tid) * 16 + tid]     = 1.0f - wx;
        sWx[(2 * tid + 1) * 16 + tid] = wx;
        // permuted K columns for GEMM2 (lets h feed B directly)
        const int k0 = (tid < 4) ? (4 * tid) : (4 * (tid - 4) + 2);
        sWy[tid * 16 + k0]     = 1.0f - wy;
        sWy[tid * 16 + k0 + 1] = wy;
    }
    __syncthreads();

    // ---- phase 2: hoist loop-invariant fragments ----
    const int lane = tid & 31;
    const int half = lane >> 4;   // 0: lanes 0-15, 1: lanes 16-31
    const int l15  = lane & 15;
    const int bi   = (int)rb;

    uint32_t goff[8];   // gather element offsets, A-fragment layout
    float    wxb[8];    // Wx B-fragments (per K-chunk: bx, by)
    float    wya[8];    // Wy A-fragments (per K-chunk: ax, ay)
    {
        const int rowoff = sRow[l15] * W_DIM;
        #pragma unroll
        for (int kc = 0; kc < 4; ++kc) {
            const int kbase = 4 * kc + 2 * half;
            goff[2 * kc]     = (uint32_t)(rowoff + sCol[kbase + 0]);
            goff[2 * kc + 1] = (uint32_t)(rowoff + sCol[kbase + 1]);
            wxb[2 * kc]      = sWx[(kbase + 0) * 16 + l15];
            wxb[2 * kc + 1]  = sWx[(kbase + 1) * 16 + l15];
            wya[2 * kc]      = sWy[l15 * 16 + kbase + 0];
            wya[2 * kc + 1]  = sWy[l15 * 16 + kbase + 1];
        }
    }

    // wave id as an explicit SGPR so slice bases stay scalar (SADDR-form loads/stores)
    const int wv = __builtin_amdgcn_readfirstlane((int)(threadIdx.x >> 5)); // 0..7
    const int s0 = part * SLICES_PER_WG + wv * ITERS;
    const float* fbase = features + (size_t)bi * SLICES * (H_DIM * W_DIM);
    float*       obase = out + (size_t)r * SLICES * (AH * AW);

    for (int it = 0; it < ITERS; ++it) {
        const int s = s0 + it;                     // slice = c*8 + t (scalar)
        const float* fb = fbase + (size_t)s * (H_DIM * W_DIM);

        if (it + 1 < ITERS) {
            __builtin_prefetch(fb + (H_DIM * W_DIM) + goff[0], 0, 0);
            __builtin_prefetch(fb + (H_DIM * W_DIM) + goff[7], 0, 0);
        }

        // gather G tile (already in A-fragment layout)
        float g[8];
        #pragma unroll
        for (int k = 0; k < 8; ++k) g[k] = fb[goff[k]];

        // GEMM1: H = G * Wx   (four v_wmma_f32_16x16x4_f32 over K=16)
        v8f h = {0.f, 0.f, 0.f, 0.f, 0.f, 0.f, 0.f, 0.f};
        #pragma unroll
        for (int kc = 0; kc < 4; ++kc) {
            v2f a; a[0] = g[2 * kc];   a[1] = g[2 * kc + 1];
            v2f b; b[0] = wxb[2 * kc]; b[1] = wxb[2 * kc + 1];
            h = __builtin_amdgcn_wmma_f32_16x16x4_f32(
                    false, a, false, b, (short)0, h, false, false);
        }

        // GEMM2: OUT = Wy_perm * H — D-layout h feeds B directly, no shuffles
        v8f o = {0.f, 0.f, 0.f, 0.f, 0.f, 0.f, 0.f, 0.f};
        #pragma unroll
        for (int kc = 0; kc < 4; ++kc) {
            v2f a; a[0] = wya[2 * kc]; a[1] = wya[2 * kc + 1];
            v2f b; b[0] = h[2 * kc];   b[1] = h[2 * kc + 1];
            o = __builtin_amdgcn_wmma_f32_16x16x4_f32(
                    false, a, false, b, (short)0, o, false, false);
        }

        // OUT rows 0..6 live in D VGPRs 0..6, lanes 0..15 (cols = lanes)
        float* dst = obase + (size_t)s * (AH * AW);
        if (lane < AW) {
            #pragma unroll
            for (int py = 0; py < AH; ++py)
                __builtin_nontemporal_store(o[py], dst + py * AW + lane);
        }
    }
}

extern "C" void kernel_launch(void* const* d_in, const int* in_sizes, int n_in,
                              void* d_out, int out_size, void* d_ws, size_t ws_size,
                              hipStream_t stream) {
    (void)n_in; (void)out_size; (void)d_ws; (void)ws_size;
    const float* features = (const float*)d_in[0];
    const float* rois     = (const float*)d_in[1];
    float*       out      = (float*)d_out;
    const int R = in_sizes[1] / 5;     // 512 ROIs
    dim3 grid(R * SPLIT);
    dim3 block(256);
    hipLaunchKernelGGL(roi_align_wmma_kernel, grid, block, 0, stream,
                       features, rois, out);
}